// GraphSAGE_LP_15126874816627
// MI455X (gfx1250) — compile-verified
//
#include <hip/hip_runtime.h>

typedef float v2f __attribute__((ext_vector_type(2)));
typedef float v8f __attribute__((ext_vector_type(8)));

#define NODE_C 128  // feature width for both scatter stages (IN_C == HID_C == 128)

// ---------------------------------------------------------------- utilities
__global__ __launch_bounds__(256) void zero_f32_kernel(float* __restrict__ p, long long n) {
    long long i = (long long)blockIdx.x * blockDim.x + threadIdx.x;
    if (i < n) p[i] = 0.0f;
}

// ---------------------------------------------------------------- degree count
__global__ __launch_bounds__(256) void degree_count_kernel(const int* __restrict__ ei,
                                                           float* __restrict__ cnt, int nedges) {
    int e = blockIdx.x * blockDim.x + threadIdx.x;
    if (e < nedges) atomicAdd(&cnt[ei[nedges + e]], 1.0f);
}

// ---------------------------------------------------------------- edge scatter
// One wave32 per edge: indices are wave-uniform -> forced scalar (s_load) via
// readfirstlane; each lane moves 4 channels with one b128 load (512B/wave
// coalesced) and 4 f32 atomics (contiguous 512B/wave).
__global__ __launch_bounds__(256) void scatter_add_kernel(const float* __restrict__ X,
                                                          const int* __restrict__ ei,
                                                          float* __restrict__ agg, int nedges) {
    int wv = (int)(((long long)blockIdx.x * blockDim.x + threadIdx.x) >> 5);
    if (wv >= nedges) return;  // uniform per wave
    int e = __builtin_amdgcn_readfirstlane(wv);
    int s = ei[e];             // edge_index[0] = src  (scalar load)
    int d = ei[nedges + e];    // edge_index[1] = dst  (scalar load)
    int lane = threadIdx.x & 31;
    const float4 v = *(const float4*)(X + (long long)s * NODE_C + lane * 4);
    float* a = agg + (long long)d * NODE_C + lane * 4;
    atomicAdd(a + 0, v.x);
    atomicAdd(a + 1, v.y);
    atomicAdd(a + 2, v.z);
    atomicAdd(a + 3, v.w);
}

// ---------------------------------------------------------------- fused SAGE GEMM
// Out[16 x NOUT] tile per block: Out = (Agg/max(cnt,1)) @ Wl + X @ Wr + bias
// K fixed at 128. One wave per 16-wide N-tile, v_wmma_f32_16x16x4_f32 over K.
// X tile staged to LDS with GLOBAL_LOAD_ASYNC_TO_LDS_B128 (ASYNCcnt path);
// mean tile staged through VGPRs (needs the 1/max(cnt,1) scale), float4-wide.
// A layout (16x4 f32): lane m=l&15; vgpr0 = A[m][k+2*(l>>4)], vgpr1 = A[m][k+2*(l>>4)+1]
// B layout (4x16 f32): lane n=l&15; vgpr0 = B[k+2*(l>>4)][n], vgpr1 = B[k+2*(l>>4)+1][n]
// C/D (16x16 f32):     vgpr j -> row (j + 8*(l>>4)), col l&15
template <int NOUT>
__global__ __launch_bounds__(32 * (NOUT / 16)) void sage_gemm_kernel(
    const float* __restrict__ Xroot, const float* __restrict__ Agg,
    const float* __restrict__ Cnt, const float* __restrict__ Wl,
    const float* __restrict__ Wr, const float* __restrict__ bias,
    float* __restrict__ Out, int nrows) {
    constexpr int K = 128;
    constexpr int LDSS = K + 4;  // 132-float stride (33x16B): bank-conflict-free A reads, 16B aligned rows
    __shared__ float Xs[16 * LDSS];
    __shared__ float Ms[16 * LDSS];

    const int m0 = blockIdx.x * 16;
    const int tid = threadIdx.x;
    constexpr int NT = 32 * (NOUT / 16);

    // stage 16 rows x 128 cols, float4 granularity (512 float4s; NT divides exactly -> full EXEC)
    for (int i = tid; i < 16 * (K / 4); i += NT) {
        int r = i >> 5;            // / (K/4)
        int c4 = (i & 31) * 4;
        int row = m0 + r;
        if (row >= nrows) row = nrows - 1;  // clamp (tail tile); stores are guarded
        // async global->LDS for the raw X tile (no transform needed)
        unsigned ldsa = (unsigned)(unsigned long long)&Xs[r * LDSS + c4];
        unsigned goff = (unsigned)(((long long)row * K + c4) * 4);
        asm volatile("global_load_async_to_lds_b128 %0, %1, %2"
                     :: "v"(ldsa), "v"(goff), "s"(Xroot) : "memory");
        // mean tile through VGPRs with per-row scale
        float inv = 1.0f / fmaxf(Cnt[row], 1.0f);
        float4 a = *(const float4*)(Agg + (long long)row * K + c4);
        a.x *= inv; a.y *= inv; a.z *= inv; a.w *= inv;
        *(float4*)(&Ms[r * LDSS + c4]) = a;
    }
    asm volatile("s_wait_asynccnt 0x0" ::: "memory");
    __syncthreads();

    const int lane = tid & 31;
    const int wave = tid >> 5;
    const int n0 = wave * 16;
    const int mn = lane & 15;            // M for A-reads, N for B-reads/stores
    const int koff = (lane >> 4) * 2;

    v8f acc = {0.f, 0.f, 0.f, 0.f, 0.f, 0.f, 0.f, 0.f};
    for (int k0 = 0; k0 < K; k0 += 4) {
        v2f ax, am, bl, br;
        ax[0] = Xs[mn * LDSS + k0 + koff];
        ax[1] = Xs[mn * LDSS + k0 + koff + 1];
        am[0] = Ms[mn * LDSS + k0 + koff];
        am[1] = Ms[mn * LDSS + k0 + koff + 1];
        bl[0] = Wl[(k0 + koff) * NOUT + n0 + mn];
        bl[1] = Wl[(k0 + koff + 1) * NOUT + n0 + mn];
        br[0] = Wr[(k0 + koff) * NOUT + n0 + mn];
        br[1] = Wr[(k0 + koff + 1) * NOUT + n0 + mn];
        acc = __builtin_amdgcn_wmma_f32_16x16x4_f32(false, am, false, bl, (short)0, acc, false, false);
        acc = __builtin_amdgcn_wmma_f32_16x16x4_f32(false, ax, false, br, (short)0, acc, false, false);
    }

    float bv = bias[n0 + mn];
    int mbase = m0 + (lane >> 4) * 8;
#pragma unroll
    for (int j = 0; j < 8; ++j) {
        int row = mbase + j;
        if (row < nrows) Out[(long long)row * NOUT + n0 + mn] = acc[j] + bv;
    }
}

// ---------------------------------------------------------------- batch-norm stats
// blockDim = 128 (thread == channel); each block strides rows -> coalesced 512B reads.
__global__ __launch_bounds__(128) void bn_stats_kernel(const float* __restrict__ H,
                                                       float* __restrict__ sum,
                                                       float* __restrict__ sumsq, int nrows) {
    int c = threadIdx.x;
    float s = 0.f, s2 = 0.f;
    for (int r = blockIdx.x; r < nrows; r += gridDim.x) {
        float v = H[(long long)r * 128 + c];
        s += v;
        s2 += v * v;
    }
    atomicAdd(&sum[c], s);
    atomicAdd(&sumsq[c], s2);
}

__global__ __launch_bounds__(128) void bn_finalize_kernel(const float* __restrict__ sum,
                                                          const float* __restrict__ sumsq,
                                                          const float* __restrict__ gamma,
                                                          const float* __restrict__ beta,
                                                          float* __restrict__ scale,
                                                          float* __restrict__ shift, int nrows) {
    int c = threadIdx.x;
    float invn = 1.0f / (float)nrows;
    float mu = sum[c] * invn;
    float var = sumsq[c] * invn - mu * mu;       // biased variance, torch BN training math
    float sc = gamma[c] * rsqrtf(var + 1e-5f);
    scale[c] = sc;
    shift[c] = beta[c] - mu * sc;
}

__global__ __launch_bounds__(256) void bn_apply_relu_kernel(float* __restrict__ H,
                                                            const float* __restrict__ scale,
                                                            const float* __restrict__ shift,
                                                            long long n) {
    long long i = (long long)blockIdx.x * blockDim.x + threadIdx.x;
    if (i >= n) return;
    int c = (int)(i & 127);
    H[i] = fmaxf(H[i] * scale[c] + shift[c], 0.0f);
}

// ---------------------------------------------------------------- label-edge dot product
// one wave32 per label edge; float2/lane (b64, 256B/wave) + xor-shuffle reduction.
__global__ __launch_bounds__(256) void edge_dot_kernel(const float* __restrict__ Z,
                                                       const int* __restrict__ eli,
                                                       float* __restrict__ out, int nlabel) {
    int w = (int)(((long long)blockIdx.x * blockDim.x + threadIdx.x) >> 5);
    int lane = threadIdx.x & 31;
    if (w >= nlabel) return;  // uniform per wave
    int wu = __builtin_amdgcn_readfirstlane(w);
    int s = eli[wu];              // scalar loads
    int d = eli[nlabel + wu];
    const float2* zs = (const float2*)(Z + (long long)s * 64);
    const float2* zd = (const float2*)(Z + (long long)d * 64);
    float2 a = zs[lane];
    float2 b = zd[lane];
    float p = a.x * b.x + a.y * b.y;
#pragma unroll
    for (int off = 16; off > 0; off >>= 1) p += __shfl_xor(p, off, 32);
    if (lane == 0) out[wu] = p;
}

// ---------------------------------------------------------------- launch
extern "C" void kernel_launch(void* const* d_in, const int* in_sizes, int n_in,
                              void* d_out, int out_size, void* d_ws, size_t ws_size,
                              hipStream_t stream) {
    const float* x     = (const float*)d_in[0];
    const int*   ei    = (const int*)d_in[1];
    const int*   eli   = (const int*)d_in[2];
    const float* W1l   = (const float*)d_in[3];
    const float* W1r   = (const float*)d_in[4];
    const float* b1    = (const float*)d_in[5];
    const float* gamma = (const float*)d_in[6];
    const float* beta  = (const float*)d_in[7];
    const float* W2l   = (const float*)d_in[8];
    const float* W2r   = (const float*)d_in[9];
    const float* b2    = (const float*)d_in[10];
    float* out = (float*)d_out;

    const int nnodes = in_sizes[0] / NODE_C;  // 50000
    const int nedges = in_sizes[1] / 2;       // 800000
    const int nlabel = in_sizes[2] / 2;       // 200000

    // workspace layout (floats)
    float* ws = (float*)d_ws;
    long long aggN = (long long)nnodes * 128;
    float* agg   = ws;                             // [N,128] scatter accumulator (reused)
    float* h     = agg + aggN;                     // [N,128] conv1 output / BN-ReLU in place
    float* z     = h + aggN;                       // [N,64]  conv2 output
    float* cnt   = z + (long long)nnodes * 64;     // [N]
    float* sum   = cnt + nnodes;                   // [128]
    float* sumsq = sum + 128;                      // [128]
    float* scale = sum + 256;                      // [128]
    float* shift = sum + 384;                      // [128]

    auto nblk = [](long long n, int b) { return (unsigned)((n + b - 1) / b); };

    // 1. zero accumulators (agg region; cnt + BN stats region)
    zero_f32_kernel<<<nblk(aggN, 256), 256, 0, stream>>>(agg, aggN);
    zero_f32_kernel<<<nblk(nnodes + 512, 256), 256, 0, stream>>>(cnt, (long long)nnodes + 512);
    // 2. in-degree counts
    degree_count_kernel<<<nblk(nedges, 256), 256, 0, stream>>>(ei, cnt, nedges);
    // 3. scatter x[src] -> agg[dst]  (one wave per edge)
    scatter_add_kernel<<<nblk((long long)nedges * 32, 256), 256, 0, stream>>>(x, ei, agg, nedges);
    // 4. h = mean @ W1l + x @ W1r + b1   (fp32 WMMA + async LDS staging)
    sage_gemm_kernel<128><<<nblk(nnodes, 16), 256, 0, stream>>>(x, agg, cnt, W1l, W1r, b1, h, nnodes);
    // 5-7. batch-norm + relu (in place on h)
    bn_stats_kernel<<<256, 128, 0, stream>>>(h, sum, sumsq, nnodes);
    bn_finalize_kernel<<<1, 128, 0, stream>>>(sum, sumsq, gamma, beta, scale, shift, nnodes);
    bn_apply_relu_kernel<<<nblk(aggN, 256), 256, 0, stream>>>(h, scale, shift, aggN);
    // 8. re-zero agg, 9. scatter h[src] -> agg[dst]
    zero_f32_kernel<<<nblk(aggN, 256), 256, 0, stream>>>(agg, aggN);
    scatter_add_kernel<<<nblk((long long)nedges * 32, 256), 256, 0, stream>>>(h, ei, agg, nedges);
    // 10. z = mean @ W2l + h @ W2r + b2
    sage_gemm_kernel<64><<<nblk(nnodes, 16), 128, 0, stream>>>(h, agg, cnt, W2l, W2r, b2, z, nnodes);
    // 11. out[e] = dot(z[src_e], z[dst_e])
    edge_dot_kernel<<<nblk((long long)nlabel * 32, 256), 256, 0, stream>>>(z, eli, out, nlabel);
}